// RelativePositionalEncodingLayer_50964081935278
// MI455X (gfx1250) — compile-verified
//
#include <hip/hip_runtime.h>
#include <hip/hip_bf16.h>

#define SEQ   2048
#define BATCH 2
#define NH    16
#define DH    64
#define SCALE 0.125f

typedef float v2f __attribute__((ext_vector_type(2)));
typedef float v4f __attribute__((ext_vector_type(4)));
typedef float v8f __attribute__((ext_vector_type(8)));

// Async DMA: 16B global -> LDS, tracked by ASYNCcnt (CDNA5).
// LDS operand = low 32 bits of the flat LDS address (= LDS byte offset).
__device__ __forceinline__ void async_copy_b128(void* lds_ptr, const void* gptr) {
  unsigned off = (unsigned)(uintptr_t)lds_ptr;
  asm volatile("global_load_async_to_lds_b128 %0, %1, off"
               :: "v"(off), "v"(gptr) : "memory");
}
__device__ __forceinline__ void wait_async_all() {
  asm volatile("s_wait_asynccnt 0x0" ::: "memory");
}

// One 16x16 (MxN) fp32 fragment of X * Y^T with K = 64, via 16 chained
// V_WMMA_F32_16X16X4_F32.  X,Y are row-major [16][stride] in LDS with K
// along the contiguous dim.  A 16x4 layout: lanes 0-15 hold M=lane,
// VGPR0/1 = K0/K1; lanes 16-31 hold K2/K3.  B 4x16 mirrors it.
__device__ __forceinline__ v8f wmma_xyt_k64(const float* __restrict__ A, int sA,
                                            const float* __restrict__ Bm, int sB,
                                            int lr, int lh, v8f c) {
#pragma unroll
  for (int kk = 0; kk < 16; ++kk) {
    int col = kk * 4 + lh * 2;
    v2f a = *(const v2f*)(A  + lr * sA + col);
    v2f b = *(const v2f*)(Bm + lr * sB + col);
    c = __builtin_amdgcn_wmma_f32_16x16x4_f32(false, a, false, b, (short)0, c,
                                              false, false);
  }
  return c;
}

// ---------------- kernel 0: sinusoidal position embedding -> ws ------------
__global__ __launch_bounds__(256) void posemb_kernel(float* __restrict__ pos_emb) {
  int idx = blockIdx.x * 256 + threadIdx.x;
  if (idx >= SEQ * 1024) return;
  int p = idx >> 10;
  int f = idx & 1023;
  float pos = (float)(SEQ - 1 - p);
  int j = (f < 512) ? f : (f - 512);
  // inv_freq[j] = 10000^(-j/512) = exp(-j * ln(10000)/512)
  float invf = __expf(-(float)j * 0.0179889460390114f);
  float ang = pos * invf;
  pos_emb[idx] = (f < 512) ? sinf(ang) : cosf(ang);
}

// ------- kernel 1: r_head_k[S,1024] = pos_emb[S,1024] @ r_net_w^T ----------
__global__ __launch_bounds__(256) void rk_gemm_kernel(const float* __restrict__ pos_emb,
                                                      const float* __restrict__ w,
                                                      float* __restrict__ rk) {
  __shared__ float sA[64 * 68];
  __shared__ float sB[64 * 68];
  int tj = blockIdx.x & 15;         // 16 tiles along output cols (1024)
  int ti = blockIdx.x >> 4;         // 32 tiles along S
  int i0 = ti * 64, o0 = tj * 64;
  int tid = threadIdx.x;
  int wave = tid >> 5, lane = tid & 31, lr = lane & 15, lh = lane >> 4;
  int f0 = wave * 2;                // 2 of 16 fragments per wave
  int fm0 = (f0 >> 2) * 16, fn0 = (f0 & 3) * 16;
  int fm1 = ((f0 + 1) >> 2) * 16, fn1 = ((f0 + 1) & 3) * 16;
  v8f acc0 = {}, acc1 = {};
  for (int kc = 0; kc < 16; ++kc) {
    __syncthreads();
    for (int idx = tid; idx < 64 * 16; idx += 256) {
      int row = idx >> 4, seg = (idx & 15) * 4;
      async_copy_b128(&sA[row * 68 + seg],
                      &pos_emb[(size_t)(i0 + row) * 1024 + kc * 64 + seg]);
      async_copy_b128(&sB[row * 68 + seg],
                      &w[(size_t)(o0 + row) * 1024 + kc * 64 + seg]);
    }
    wait_async_all();
    __syncthreads();
    acc0 = wmma_xyt_k64(&sA[fm0 * 68], 68, &sB[fn0 * 68], 68, lr, lh, acc0);
    acc1 = wmma_xyt_k64(&sA[fm1 * 68], 68, &sB[fn1 * 68], 68, lr, lh, acc1);
  }
#pragma unroll
  for (int vg = 0; vg < 8; ++vg) {
    int m = fm0 + vg + lh * 8, n = fn0 + lr;
    rk[(size_t)(i0 + m) * 1024 + o0 + n] = acc0[vg];
    m = fm1 + vg + lh * 8; n = fn1 + lr;
    rk[(size_t)(i0 + m) * 1024 + o0 + n] = acc1[vg];
  }
}

// ---------------- kernel 2: fused AC + shifted BD attention ----------------
// out[b,h,i,j] = SCALE * ( (q[i]+rw)·k[j] + P[i + (v>=0)][v] ), v = j-i-2,
// P[r][v] = (q[r]+rr)·Rkv[v],  Rkv[-1]=0, Rkv[v<0]=Rk[v+S+1], Rkv[v>=0]=Rk[v]
__global__ __launch_bounds__(256) void relattn_kernel(const float* __restrict__ q,
                                                      const float* __restrict__ kmat,
                                                      const float* __restrict__ rw,
                                                      const float* __restrict__ rr,
                                                      const float* __restrict__ rk,
                                                      float* __restrict__ out) {
  __shared__ float s_qrw[64 * 68];    // q + r_w_bias, rows i0..i0+63
  __shared__ float s_rrq[80 * 68];    // q + r_r_bias, rows i0..i0+79
  __shared__ float s_kt [64 * 68];    // k rows j0..j0+63
  __shared__ float s_rkv[144 * 68];   // Rkv virtual rows v0..v0+143
  __shared__ float s_p  [80 * 145];   // P tile
  __shared__ float s_o  [64 * 64];    // output staging

  int blk = blockIdx.x;
  int tj = blk & 31, ti = (blk >> 5) & 31, h = (blk >> 10) & 15, b = blk >> 14;
  int i0 = ti * 64, j0 = tj * 64, diag = j0 - i0;
  int tid = threadIdx.x, wave = tid >> 5, lane = tid & 31;
  int lr = lane & 15, lh = lane >> 4;

  // ---- stage k tile: pure copy via async DMA to LDS ----
  for (int idx = tid; idx < 64 * 16; idx += 256) {
    int row = idx >> 4, seg = (idx & 15) * 4;
    int gj = j0 + row;
    async_copy_b128(&s_kt[row * 68 + seg],
                    &kmat[(((size_t)gj * BATCH + b) * NH + h) * DH + seg]);
  }
  // ---- stage Rkv virtual rows (v0 = diag - 65): async where valid ----
  for (int idx = tid; idx < 144 * 16; idx += 256) {
    int row = idx >> 4, seg = (idx & 15) * 4;
    int vg = diag - 65 + row;
    int rrow = (vg >= 0) ? vg : (vg + SEQ + 1);
    bool valid = (vg != -1) && (rrow >= 0) && (rrow < SEQ);
    if (valid) {
      async_copy_b128(&s_rkv[row * 68 + seg],
                      &rk[(size_t)rrow * 1024 + h * DH + seg]);
    } else {
      v4f z = {0.f, 0.f, 0.f, 0.f};
      *(v4f*)&s_rkv[row * 68 + seg] = z;
    }
  }
  // ---- stage q tile with both biases (80 rows, clamp past S to zero) ----
  for (int idx = tid; idx < 80 * 16; idx += 256) {
    int row = idx >> 4, seg = (idx & 15) * 4;
    int gi = i0 + row;
    v4f qv = {0.f, 0.f, 0.f, 0.f};
    if (gi < SEQ)
      qv = *(const v4f*)&q[(((size_t)gi * BATCH + b) * NH + h) * DH + seg];
    v4f rwv = *(const v4f*)&rw[h * DH + seg];
    v4f rrv = *(const v4f*)&rr[h * DH + seg];
    if (row < 64) *(v4f*)&s_qrw[row * 68 + seg] = qv + rwv;
    *(v4f*)&s_rrq[row * 68 + seg] = qv + rrv;
  }
  wait_async_all();
  __syncthreads();

  // ---- AC: 4x4 fragments, 2 per wave (kept in registers) ----
  int fA0 = wave * 2, fA1 = wave * 2 + 1;
  int am0 = (fA0 >> 2) * 16, an0 = (fA0 & 3) * 16;
  int am1 = (fA1 >> 2) * 16, an1 = (fA1 & 3) * 16;
  v8f ac0 = {}, ac1 = {};
  ac0 = wmma_xyt_k64(&s_qrw[am0 * 68], 68, &s_kt[an0 * 68], 68, lr, lh, ac0);
  ac1 = wmma_xyt_k64(&s_qrw[am1 * 68], 68, &s_kt[an1 * 68], 68, lr, lh, ac1);

  // ---- P: 5x9 fragments round-robin over waves, store to LDS ----
  for (int f = wave; f < 45; f += 8) {
    int pm = (f / 9) * 16, pn = (f % 9) * 16;
    v8f pc = {};
    pc = wmma_xyt_k64(&s_rrq[pm * 68], 68, &s_rkv[pn * 68], 68, lr, lh, pc);
#pragma unroll
    for (int vg = 0; vg < 8; ++vg) {
      int m = pm + vg + lh * 8, n = pn + lr;
      s_p[m * 145 + n] = pc[vg];
    }
  }
  __syncthreads();

  // ---- combine: out = SCALE*(AC + P[m + (v>=0)][(n-m)+63]) ----
#pragma unroll
  for (int vg = 0; vg < 8; ++vg) {
    {
      int m = am0 + vg + lh * 8, n = an0 + lr;
      int dmn = n - m;
      int vv = diag + dmn - 2;
      int r = m + (vv >= 0 ? 1 : 0);
      s_o[m * 64 + n] = SCALE * (ac0[vg] + s_p[r * 145 + dmn + 63]);
    }
    {
      int m = am1 + vg + lh * 8, n = an1 + lr;
      int dmn = n - m;
      int vv = diag + dmn - 2;
      int r = m + (vv >= 0 ? 1 : 0);
      s_o[m * 64 + n] = SCALE * (ac1[vg] + s_p[r * 145 + dmn + 63]);
    }
  }
  __syncthreads();

  // ---- stream tile out: coalesced b128 non-temporal stores ----
  size_t obase = ((size_t)(b * NH + h) * SEQ + i0) * SEQ + j0;
  for (int idx = tid; idx < 64 * 16; idx += 256) {
    int row = idx >> 4, seg = (idx & 15) * 4;
    v4f v = *(const v4f*)&s_o[row * 64 + seg];
    __builtin_nontemporal_store(v, (v4f*)&out[obase + (size_t)row * SEQ + seg]);
  }
}

extern "C" void kernel_launch(void* const* d_in, const int* in_sizes, int n_in,
                              void* d_out, int out_size, void* d_ws, size_t ws_size,
                              hipStream_t stream) {
  const float* q  = (const float*)d_in[0];  // [S,B,H,D]
  const float* k  = (const float*)d_in[1];  // [S,B,H,D]
  const float* rw = (const float*)d_in[2];  // [H,D]
  const float* rr = (const float*)d_in[3];  // [H,D]
  const float* w  = (const float*)d_in[4];  // [1024,1024]
  float* out = (float*)d_out;               // [B,H,S,S]

  float* pos = (float*)d_ws;                // [S,1024]  8 MB
  float* rk  = pos + (size_t)SEQ * 1024;    // [S,1024]  8 MB

  posemb_kernel<<<(SEQ * 1024) / 256, 256, 0, stream>>>(pos);
  rk_gemm_kernel<<<32 * 16, 256, 0, stream>>>(pos, w, rk);
  relattn_kernel<<<BATCH * NH * 32 * 32, 256, 0, stream>>>(q, k, rw, rr, rk, out);
}